// Sampler_38680475468117
// MI455X (gfx1250) — compile-verified
//
#include <hip/hip_runtime.h>
#include <hip/hip_bf16.h>
#include <math.h>

#define B_ROWS   128
#define VOCAB    128000
#define NTHREADS 512          // 16 waves of 32 on gfx1250
#define NB1      4096         // coarse bins over x = (l - max)/T, range [-64, 0]
#define NB2      2048         // fine sub-bins per coarse bin (2^23 total resolution)
#define LOG2E    1.4426950408889634f

typedef __attribute__((ext_vector_type(2))) float v2f;
typedef __attribute__((ext_vector_type(8))) float v8f;

__device__ __forceinline__ float fast_e(float x) { return __builtin_exp2f(x * LOG2E); }

// Map xn = -x >= 0 to (coarse bin, fine sub-bin). Bin 0 holds the largest x.
__device__ __forceinline__ void bin_of(float xn, int& bin, int& sub) {
    float t = xn * 64.0f;                   // coarse width = 1/64
    int b = (int)t;
    if (b < 0) b = 0;
    if (b > NB1 - 1) b = NB1 - 1;
    float fr = t - (float)b;
    int s = (int)(fr * (float)NB2);
    if (s < 0) s = 0;
    if (s > NB2 - 1) s = NB2 - 1;
    bin = b; sub = s;
}

// ---------------------------------------------------------------------------
// 256-element inclusive scan per wave via V_WMMA_F32_16X16X4_F32.
// View chunk as 16x16 row-major M.  RowScan = M x U (U upper-tri ones,
// 4 chained K=4 WMMAs).  Row offsets = L x t (strict-lower-tri ones times
// row totals, 4 more WMMAs).  Final = RowScan + Offsets elementwise (the
// broadcast-over-columns D layout lines up with the C/D VGPR mapping).
// A 16x4 f32 operand layout: lane L holds (M = L%16, K = v + 2*(L/16)).
// ---------------------------------------------------------------------------
template <typename S>
__device__ __forceinline__ void wmma_scan_chunk(const S* __restrict__ src,
                                                float* __restrict__ dst,
                                                float* t16, float* total) {
    const int lane = threadIdx.x & 31;
    const int hi   = lane >> 4;
    const int mrow = lane & 15;

    v8f C = {0.f,0.f,0.f,0.f,0.f,0.f,0.f,0.f};
#pragma unroll
    for (int j = 0; j < 4; ++j) {
        const int k0 = 4 * j + 2 * hi;
        v2f a, b;
        a[0] = (float)src[mrow * 16 + k0];
        a[1] = (float)src[mrow * 16 + k0 + 1];
        b[0] = (k0     <= mrow) ? 1.0f : 0.0f;   // U[k][n] = (k <= n)
        b[1] = (k0 + 1 <= mrow) ? 1.0f : 0.0f;
        C = __builtin_amdgcn_wmma_f32_16x16x4_f32(false, a, false, b,
                                                  (short)0, C, false, false);
    }
    // Row totals t_r = RowScan[r][15] live in lanes 15 (rows 0-7) / 31 (rows 8-15)
    if (mrow == 15) {
#pragma unroll
        for (int g = 0; g < 8; ++g) t16[g + 8 * hi] = C[g];
    }
    asm volatile("s_wait_dscnt 0" ::: "memory");   // in-wave LDS RAW fence

    v8f O = {0.f,0.f,0.f,0.f,0.f,0.f,0.f,0.f};
#pragma unroll
    for (int j = 0; j < 4; ++j) {
        const int k0 = 4 * j + 2 * hi;
        v2f a, b;
        a[0] = (k0     < mrow) ? 1.0f : 0.0f;    // L[i][k] = (k < i)
        a[1] = (k0 + 1 < mrow) ? 1.0f : 0.0f;
        b[0] = t16[k0];                          // t broadcast over N
        b[1] = t16[k0 + 1];
        O = __builtin_amdgcn_wmma_f32_16x16x4_f32(false, a, false, b,
                                                  (short)0, O, false, false);
    }
    // C/D layout: VGPR g, lanes<16 -> row g, lanes>=16 -> row g+8; col = lane%16
#pragma unroll
    for (int g = 0; g < 8; ++g)
        dst[(g + 8 * hi) * 16 + mrow] = C[g] + O[g];
    if (lane == 31) *total = C[7] + O[7];        // element (15,15) = chunk sum
}

// Exclusive ("mass strictly before") scan of N (256-multiple) elements in LDS.
template <typename S>
__device__ void scan_exclusive(const S* src, float* dst, int N,
                               float* tbuf, float* ctot) {
    const int w   = threadIdx.x >> 5;
    const int nch = N >> 8;
    if (w < nch)   // wave-uniform branch: EXEC all ones inside (WMMA-legal)
        wmma_scan_chunk(src + (w << 8), dst + (w << 8), tbuf + w * 16, ctot + w);
    __syncthreads();
    if (threadIdx.x == 0) {
        float run = 0.f;
        for (int c = 0; c < nch; ++c) { float t = ctot[c]; ctot[c] = run; run += t; }
    }
    __syncthreads();
    for (int i = threadIdx.x; i < N; i += NTHREADS)
        dst[i] = dst[i] - (float)src[i] + ctot[i >> 8];
    __syncthreads();
}

__global__ __launch_bounds__(NTHREADS)
void Sampler_38680475468117_kernel(const float* __restrict__ logits,
                                   const float* __restrict__ temps,
                                   const float* __restrict__ tops,
                                   const float* __restrict__ minps,
                                   const float* __restrict__ us,
                                   const int*   __restrict__ topks,
                                   float* __restrict__ out) {
    extern __shared__ __align__(16) char smem[];
    float*    hmass = (float*)smem;                  // [NB1] coarse mass
    float*    cmass = hmass + NB1;                   // [NB1] exclusive mass scan
    unsigned* hcnt  = (unsigned*)(cmass + NB1);      // [NB1] coarse count
    float*    ccnt  = (float*)(hcnt + NB1);          // [NB1] exclusive count scan
    unsigned* fcnt  = (unsigned*)(ccnt + NB1);       // [NB2] fine count  (bin b_k)
    float*    fcs   = (float*)(fcnt + NB2);
    float*    fmk   = fcs  + NB2;                    // fine mass (b_k / reused b_s)
    float*    fmks  = fmk  + NB2;
    float*    fmp   = fmks + NB2;                    // fine mass (b_p)
    float*    fmps  = fmp  + NB2;
    float*    fmm   = fmps + NB2;                    // fine mass (b_m)
    float*    fmms  = fmm  + NB2;
    float*    tbuf  = fmms + NB2;                    // [16*16] WMMA t scratch
    float*    ctot  = tbuf + 256;                    // [16] chunk totals
    float*    red   = ctot + 16;                     // [NTHREADS] reductions
    int*      sint  = (int*)(red + NTHREADS);        // 0:b_k 1:b_p 2:b_s 3:s_k 4:s_p 5:s_s
    unsigned long long* skey = (unsigned long long*)(sint + 8);

    const int tid = threadIdx.x;
    const int row = blockIdx.x;
    const float* Lp = logits + (size_t)row * VOCAB;
    const float invT = 1.0f / temps[row];

    // ---- P1: row max ----
    float mx = -INFINITY;
    for (int i = tid; i < VOCAB; i += NTHREADS) {
        if (i + 4 * NTHREADS < VOCAB)
            __builtin_prefetch(&Lp[i + 4 * NTHREADS], 0, 0);   // global_prefetch_b8
        mx = fmaxf(mx, Lp[i]);
    }
    red[tid] = mx;
    __syncthreads();
    for (int s = NTHREADS >> 1; s > 0; s >>= 1) {
        if (tid < s) red[tid] = fmaxf(red[tid], red[tid + s]);
        __syncthreads();
    }
    const float m = red[0];
    __syncthreads();

    // init hists + search defaults
    for (int i = tid; i < NB1; i += NTHREADS) { hcnt[i] = 0u; hmass[i] = 0.f; }
    if (tid < 8) sint[tid] = (tid < 3) ? (NB1 - 1) : (NB2 - 1);
    if (tid == 0) *skey = 0ull;
    __syncthreads();

    // ---- P2: coarse histogram (count + unnormalized prob mass) ----
    for (int i = tid; i < VOCAB; i += NTHREADS) {
        float x = (Lp[i] - m) * invT;
        float e = fast_e(x);
        int bin, sub; bin_of(-x, bin, sub); (void)sub;
        atomicAdd(&hcnt[bin], 1u);          // ds_add_u32
        atomicAdd(&hmass[bin], e);          // ds_add_f32
    }
    __syncthreads();
    scan_exclusive(hcnt,  ccnt,  NB1, tbuf, ctot);
    scan_exclusive(hmass, cmass, NB1, tbuf, ctot);

    // Totals & thresholds (recomputed per-thread from shared arrays)
    const float Z  = cmass[NB1 - 1] + hmass[NB1 - 1];   // hist total == softmax Z
    const float Pp = tops[row] * Z;                      // top-p mass budget
    const float Kf = (float)topks[row];
    const float mp = minps[row];
    const bool  mp_all = !(mp > 1e-28f);
    int b_m = NB1 - 1, s_m = NB2 - 1;                    // min-p cut: x >= ln(min_p)
    if (!mp_all) { float xm = -__builtin_logf(mp); if (xm > 0.f) bin_of(xm, b_m, s_m); else { b_m = 0; s_m = 0; } }

    // ---- coarse boundary searches (unique crossing bin each) ----
    for (int i = tid; i < NB1; i += NTHREADS) {
        float cb = ccnt[i],  cv = (float)hcnt[i];
        if (cb < Kf && cb + cv >= Kf) atomicMin(&sint[0], i);        // top-k bin
        float mb = cmass[i], mv = hmass[i];
        if (mb <= Pp && mb + mv > Pp) atomicMin(&sint[1], i);        // top-p bin
    }
    __syncthreads();
    const int b_k = sint[0];
    const int b_p = sint[1];

    // ---- P3: fine histograms of the boundary bins ----
    for (int i = tid; i < NB2; i += NTHREADS) { fcnt[i] = 0u; fmk[i] = 0.f; fmp[i] = 0.f; fmm[i] = 0.f; }
    __syncthreads();
    for (int i = tid; i < VOCAB; i += NTHREADS) {
        float x = (Lp[i] - m) * invT;
        float e = fast_e(x);
        int bin, sub; bin_of(-x, bin, sub);
        if (bin == b_k) { atomicAdd(&fcnt[sub], 1u); atomicAdd(&fmk[sub], e); }
        if (bin == b_p) atomicAdd(&fmp[sub], e);
        if (bin == b_m) atomicAdd(&fmm[sub], e);
    }
    __syncthreads();
    scan_exclusive(fcnt, fcs,  NB2, tbuf, ctot);
    scan_exclusive(fmk,  fmks, NB2, tbuf, ctot);
    scan_exclusive(fmp,  fmps, NB2, tbuf, ctot);
    scan_exclusive(fmm,  fmms, NB2, tbuf, ctot);

    const float remK = Kf - ccnt[b_k];
    const float remP = Pp - cmass[b_p];
    for (int s = tid; s < NB2; s += NTHREADS) {
        float cb = fcs[s],  cv = (float)fcnt[s];
        if (cb < remK && cb + cv >= remK) atomicMin(&sint[3], s);
        float mb = fmps[s], mv = fmp[s];
        if (mb <= remP && mb + mv > remP) atomicMin(&sint[4], s);
    }
    __syncthreads();
    const int s_k = sint[3], s_p = sint[4];

    // Kept masses at each boundary; sampling kept set = lex-min boundary
    const float Mk = cmass[b_k] + fmks[s_k] + fmk[s_k];
    const float Mp = cmass[b_p] + fmps[s_p] + fmp[s_p];
    const float Mm = mp_all ? Z : (cmass[b_m] + fmms[s_m] + fmm[s_m]);
    const float Mkept = fminf(Mk, fminf(Mp, Mm));
    const float tau   = us[row] * Mkept;                 // inverse-CDF target

    // ---- locate sampled item: coarse bin ----
    for (int i = tid; i < NB1; i += NTHREADS) {
        float mb = cmass[i], mv = hmass[i];
        if (mb <= tau && mb + mv > tau) atomicMin(&sint[2], i);
    }
    __syncthreads();
    const int b_s = sint[2];

    const float* fsv; const float* fss;
    if (b_s == b_k)              { fsv = fmk; fss = fmks; }
    else if (b_s == b_p)         { fsv = fmp; fss = fmps; }
    else if (b_s == b_m && !mp_all) { fsv = fmm; fss = fmms; }
    else {                       // rare: refine b_s (uniform branch)
        for (int i = tid; i < NB2; i += NTHREADS) fmk[i] = 0.f;
        __syncthreads();
        for (int i = tid; i < VOCAB; i += NTHREADS) {
            float x = (Lp[i] - m) * invT;
            float e = fast_e(x);
            int bin, sub; bin_of(-x, bin, sub);
            if (bin == b_s) atomicAdd(&fmk[sub], e);
        }
        __syncthreads();
        scan_exclusive(fmk, fmks, NB2, tbuf, ctot);
        fsv = fmk; fss = fmks;
    }
    const float remS = tau - cmass[b_s];
    for (int s = tid; s < NB2; s += NTHREADS) {
        float mb = fss[s], mv = fsv[s];
        if (mb <= remS && mb + mv > remS) atomicMin(&sint[5], s);
    }
    __syncthreads();
    const int s_s = sint[5];

    // ---- P5: recover token id in (b_s, s_s): max e, tie -> min index ----
    for (int i = tid; i < VOCAB; i += NTHREADS) {
        float x = (Lp[i] - m) * invT;
        float e = fast_e(x);
        int bin, sub; bin_of(-x, bin, sub);
        if (bin == b_s && sub == s_s) {
            unsigned long long key =
                ((unsigned long long)__float_as_uint(e) << 32) |
                (unsigned long long)(0x7FFFFFFFu - (unsigned)i);
            atomicMax(skey, key);
        }
    }
    __syncthreads();
    unsigned long long key = *skey;
    int token = (key == 0ull) ? 0 : (int)(0x7FFFFFFFu - (unsigned)(key & 0xFFFFFFFFull));
    if (token < 0 || token >= VOCAB) token = 0;

    // ---- P6: top-p normalized logprobs (log e_i == x_i analytically) ----
    const float logMp = __builtin_logf(Mp);
    float* lp_out = out + 2 * B_ROWS + (size_t)row * VOCAB;
    for (int i = tid; i < VOCAB; i += NTHREADS) {
        float x = (Lp[i] - m) * invT;
        int bin, sub; bin_of(-x, bin, sub);
        bool kept = (bin < b_p) || (bin == b_p && sub <= s_p);
        float v = kept ? (x - logMp) : -INFINITY;
        lp_out[i] = v;
        if (i == token) out[B_ROWS + row] = v;   // next_token_logprob
    }
    if (tid == 0) out[row] = (float)token;       // token id (exact in f32)
}

extern "C" void kernel_launch(void* const* d_in, const int* in_sizes, int n_in,
                              void* d_out, int out_size, void* d_ws, size_t ws_size,
                              hipStream_t stream) {
    const float* logits = (const float*)d_in[0];
    const float* temps  = (const float*)d_in[1];
    const float* tops   = (const float*)d_in[2];
    const float* minps  = (const float*)d_in[3];
    const float* us     = (const float*)d_in[4];
    const int*   topks  = (const int*)d_in[5];
    float* out = (float*)d_out;
    (void)in_sizes; (void)n_in; (void)out_size; (void)d_ws; (void)ws_size;

    const size_t lds = (size_t)(4 * NB1 + 8 * NB2 + 256 + 16 + NTHREADS) * sizeof(float)
                     + 8 * sizeof(int) + sizeof(unsigned long long);
    Sampler_38680475468117_kernel<<<dim3(B_ROWS), dim3(NTHREADS), lds, stream>>>(
        logits, temps, tops, minps, us, topks, out);
}